// DispersionD3_90117003805375
// MI455X (gfx1250) — compile-verified
//
#include <hip/hip_runtime.h>
#include <cstdint>

// ---------------------------------------------------------------------------
// DFT-D3 dispersion energy, CDNA5 (gfx1250, wave32).
// Pass 1: coordination numbers (vec4 streaming + global_atomic_add_f32).
// Pass 2: 25-term Gaussian C6 interpolation + BJ damping.
//   - Tables async-DMA'd global->LDS (global_load_async_to_lds_b32 +
//     s_wait_asynccnt), packed into a float4 LDS table -> ONE ds_load_b128
//     per term, bank-conflict-free across the 16 species combos.
//   - Scratch interleaves {cn, species} as float2 so per-pair random gathers
//     are 2x global_load_b64 instead of 4x b32 (when ws_size allows).
//   - Per-molecule segment sum via ds_add_f32, one global atomic flush/block.
// ---------------------------------------------------------------------------

#define DEVINL __device__ __forceinline__

namespace {
constexpr float ANG2BOHR = 1.8897261258369282f;
constexpr float LOG2E    = 1.4426950408889634f;
constexpr float K1       = 16.0f;
constexpr float K2       = 4.0f / 3.0f;
constexpr float K3       = 4.0f;
constexpr float EPS      = 1e-35f;
constexpr int   NE       = 4;          // elements
constexpr int   NC       = NE * NE;    // species combos
constexpr int   GG       = 25;         // G*G gaussian table entries
}

#if __has_builtin(__builtin_amdgcn_global_load_async_to_lds_b32) && \
    __has_builtin(__builtin_amdgcn_s_wait_asynccnt)
#define HAVE_ASYNC_LDS 1
#else
#define HAVE_ASYNC_LDS 0
#endif

typedef __attribute__((address_space(1))) int* gptr_i32;
typedef __attribute__((address_space(3))) int* lptr_i32;

// Stage n floats from global g into LDS l, cooperatively across the block.
// Thread t handles elements t, t+blockDim, ... (pack step reuses this map).
DEVINL void stage_to_lds(const float* __restrict__ g, float* l, int n) {
#if HAVE_ASYNC_LDS
  for (int t = threadIdx.x; t < n; t += blockDim.x) {
    __builtin_amdgcn_global_load_async_to_lds_b32(
        (gptr_i32)(g + t), (lptr_i32)(l + t), /*offset=*/0, /*cpol=*/0);
  }
#else
  for (int t = threadIdx.x; t < n; t += blockDim.x) l[t] = g[t];
#endif
}

DEVINL void wait_async_stage() {
#if HAVE_ASYNC_LDS
  __builtin_amdgcn_s_wait_asynccnt(0);
#endif
}

// ---------------------------------------------------------------------------
// Scratch layout helpers.  ILV=true: d_ws holds float2{cn, bitcast(species)}.
// ILV=false: d_ws holds float cn[] only; species read from its own array.
// ---------------------------------------------------------------------------
template <bool ILV>
DEVINL float* cn_slot(float* ws, int i) {
  return ILV ? (ws + 2 * i) : (ws + i);
}

// ---------------------------------------------------------------------------
// Kernel 0: init scratch (cn=0, optionally species in .y) and zero output.
// ---------------------------------------------------------------------------
template <bool ILV>
__global__ void k_init(float* __restrict__ ws, int ncn,
                       const int* __restrict__ species,
                       float* __restrict__ out, int nout) {
  int i = blockIdx.x * blockDim.x + threadIdx.x;
  if (i < ncn) {
    if (ILV) {
      ((float2*)ws)[i] = make_float2(0.0f, __int_as_float(species[i]));
    } else {
      ws[i] = 0.0f;
    }
  }
  if (i < nout) out[i] = 0.0f;
}

// ---------------------------------------------------------------------------
// Kernel 1: coordination numbers.
//   counting = 1 / (1 + exp(-K1 * (K2 * rsum / d - 1)))
// ---------------------------------------------------------------------------
template <bool ILV>
DEVINL void cn_pair(int i0, int i1, float dA,
                    const int* __restrict__ species,
                    const float* __restrict__ sCov,
                    float* __restrict__ ws) {
  float d    = dA * ANG2BOHR;
  float rsum = sCov[species[i0]] + sCov[species[i1]];
  float y    = K2 * rsum * __builtin_amdgcn_rcpf(d) - 1.0f;
  // exp(-K1*y) == exp2(-K1*log2(e)*y)  -> single v_exp_f32
  float e    = __builtin_amdgcn_exp2f(y * (-K1 * LOG2E));
  float counting = __builtin_amdgcn_rcpf(1.0f + e);
  atomicAdd(cn_slot<ILV>(ws, i0), counting);
  atomicAdd(cn_slot<ILV>(ws, i1), counting);
}

template <bool ILV>
__global__ void __launch_bounds__(256)
k_coordnum(const int* __restrict__ idx,        // [2*P]
           const float* __restrict__ dist,     // [P] (angstrom)
           const int* __restrict__ species,    // [M*A] flattened
           const float* __restrict__ covalent, // [NE]
           float* __restrict__ ws,             // cn scratch (layout per ILV)
           int P) {
  __shared__ float sCov[NE];
  if (threadIdx.x < NE) sCov[threadIdx.x] = covalent[threadIdx.x];
  __syncthreads();

  const int* __restrict__ i0s = idx;
  const int* __restrict__ i1s = idx + P;
  const int gtid   = blockIdx.x * blockDim.x + threadIdx.x;
  const int stride = gridDim.x * blockDim.x;

  if ((P & 3) == 0) {
    // vec4 streaming: global_load_b128 for indices and distances
    const int     nv  = P >> 2;
    const int4*   i04 = (const int4*)i0s;
    const int4*   i14 = (const int4*)i1s;
    const float4* d4  = (const float4*)dist;
    for (int v = gtid; v < nv; v += stride) {
      if (v + stride < nv) {
        __builtin_prefetch(&i04[v + stride], 0, 0);
        __builtin_prefetch(&i14[v + stride], 0, 0);
        __builtin_prefetch(&d4[v + stride], 0, 0);
      }
      int4   a  = i04[v];
      int4   b  = i14[v];
      float4 dd = d4[v];
      cn_pair<ILV>(a.x, b.x, dd.x, species, sCov, ws);
      cn_pair<ILV>(a.y, b.y, dd.y, species, sCov, ws);
      cn_pair<ILV>(a.z, b.z, dd.z, species, sCov, ws);
      cn_pair<ILV>(a.w, b.w, dd.w, species, sCov, ws);
    }
  } else {
    for (int p = gtid; p < P; p += stride)
      cn_pair<ILV>(i0s[p], i1s[p], dist[p], species, sCov, ws);
  }
}

// ---------------------------------------------------------------------------
// Kernel 2: pair energies + per-molecule segment sum.
// LDS: float4 tab[400] | raw c6[400] raw cnA[400] raw cnB[400] |
//      q[16] d6[16] d8[16] | mol[M]
// ---------------------------------------------------------------------------
template <bool ILV>
__global__ void __launch_bounds__(256)
k_energy(const int* __restrict__ idx,          // [2*P]
         const float* __restrict__ dist,       // [P]
         const int* __restrict__ species,      // [M*A]
         const float* __restrict__ ws,         // cn scratch (layout per ILV)
         const float* __restrict__ c6tab,      // [NC*GG]
         const float* __restrict__ cnA,        // [NC*GG]
         const float* __restrict__ cnB,        // [NC*GG]
         const float* __restrict__ cutoff,     // [NC]
         const float* __restrict__ sqq,        // [NC]
         const float* __restrict__ a1p, const float* __restrict__ a2p,
         const float* __restrict__ s6p, const float* __restrict__ s8p,
         float* __restrict__ energies,         // [M]
         int P, unsigned A, int molShift, int M) {
  extern __shared__ float4 smem4[];
  float4* sTab = smem4;                       // NC*GG float4 (16B aligned)
  float* smem  = (float*)(smem4 + NC * GG);
  float* rawC6 = smem;                        // NC*GG
  float* rawA  = smem + NC * GG;              // NC*GG
  float* rawB  = smem + 2 * NC * GG;          // NC*GG
  float* sQ    = smem + 3 * NC * GG;          // NC
  float* sD6   = smem + 3 * NC * GG + NC;     // NC
  float* sD8   = smem + 3 * NC * GG + 2*NC;   // NC
  float* sMol  = smem + 3 * NC * GG + 3*NC;   // M

  // async DMA the 3x400-entry gaussian tables into LDS
  stage_to_lds(c6tab, rawC6, NC * GG);
  stage_to_lds(cnA,   rawA,  NC * GG);
  stage_to_lds(cnB,   rawB,  NC * GG);

  const float a1v = *a1p, a2v = *a2p, s6v = *s6p, s8v = *s8p;
  if (threadIdx.x < NC) {
    int c = threadIdx.x;
    float damp  = a1v * cutoff[c] + a2v;
    float damp2 = damp * damp;
    float damp4 = damp2 * damp2;
    sQ[c]  = sqq[c];
    sD6[c] = damp4 * damp2;
    sD8[c] = damp4 * damp4;
  }
  for (int m = threadIdx.x; m < M; m += blockDim.x) sMol[m] = 0.0f;

  wait_async_stage();   // this wave's async copies are now in LDS
  // Pack (cnA, cnB, c6, mask) -> float4; same thread->element striding as the
  // staging loops, so each thread packs exactly the elements it staged.
  for (int e = threadIdx.x; e < NC * GG; e += blockDim.x) {
    float c6v = rawC6[e];
    sTab[e] = make_float4(rawA[e], rawB[e], c6v, c6v <= 0.0f ? 0.0f : 1.0f);
  }
  __syncthreads();      // all waves' LDS writes visible

  const int* __restrict__ i0s = idx;
  const int* __restrict__ i1s = idx + P;
  const int stride = gridDim.x * blockDim.x;

  for (int p = blockIdx.x * blockDim.x + threadIdx.x; p < P; p += stride) {
    if (p + stride < P) {
      __builtin_prefetch(&i0s[p + stride], 0, 0);
      __builtin_prefetch(&i1s[p + stride], 0, 0);
      __builtin_prefetch(&dist[p + stride], 0, 0);
    }
    int   i0 = i0s[p];
    int   i1 = i1s[p];
    float d  = dist[p] * ANG2BOHR;

    float cn0, cn1;
    int s0, s1;
    if (ILV) {
      // single global_load_b64 gather per pair member: {cn, species}
      float2 q0 = ((const float2*)ws)[i0];
      float2 q1 = ((const float2*)ws)[i1];
      cn0 = q0.x; s0 = __float_as_int(q0.y);
      cn1 = q1.x; s1 = __float_as_int(q1.y);
    } else {
      cn0 = ws[i0]; cn1 = ws[i1];
      s0 = species[i0]; s1 = species[i1];
    }
    int c = s0 * NE + s1;

    const float4* t4 = sTab + c * GG;  // combo stride 400B: conflict-free

    float w = EPS, z = EPS;
#pragma unroll
    for (int j = 0; j < GG; ++j) {
      float4 q = t4[j];                // one ds_load_b128
      float dx = cn0 - q.x;
      float dy = cn1 - q.y;
      float t  = fmaf(dx, dx, dy * dy);
      // exp(-K3*t) == exp2(-K3*log2(e)*t) -> v_exp_f32; q.w is the 0/1 mask
      float g  = __builtin_amdgcn_exp2f(t * (-K3 * LOG2E)) * q.w;
      w += g;
      z = fmaf(q.z, g, z);
    }
    float c6 = z * __builtin_amdgcn_rcpf(w);
    float c8 = 3.0f * c6 * sQ[c];

    float d2 = d * d;
    float d4 = d2 * d2;
    float d6 = d4 * d2;
    float d8 = d4 * d4;
    float e  = -(s6v * c6 * __builtin_amdgcn_rcpf(d6 + sD6[c]) +
                 s8v * c8 * __builtin_amdgcn_rcpf(d8 + sD8[c]));

    unsigned mol = (molShift >= 0) ? ((unsigned)i0 >> molShift)
                                   : ((unsigned)i0 / A);
    atomicAdd(&sMol[mol], e);          // ds_add_f32
  }

  __syncthreads();
  for (int m = threadIdx.x; m < M; m += blockDim.x) {
    float v = sMol[m];
    if (v != 0.0f) atomicAdd(&energies[m], v);  // global_atomic_add_f32
  }
}

// ---------------------------------------------------------------------------
// Host launch
// ---------------------------------------------------------------------------
extern "C" void kernel_launch(void* const* d_in, const int* in_sizes, int n_in,
                              void* d_out, int out_size, void* d_ws, size_t ws_size,
                              hipStream_t stream) {
  const int*   species  = (const int*)  d_in[0];
  const int*   idx      = (const int*)  d_in[1];
  const float* dist     = (const float*)d_in[2];
  const float* cutoff   = (const float*)d_in[3];
  const float* c6tab    = (const float*)d_in[4];
  const float* cnA      = (const float*)d_in[5];
  const float* cnB      = (const float*)d_in[6];
  const float* covalent = (const float*)d_in[7];
  const float* sqq      = (const float*)d_in[8];
  const float* a1       = (const float*)d_in[9];
  const float* a2       = (const float*)d_in[10];
  const float* s6       = (const float*)d_in[11];
  const float* s8       = (const float*)d_in[12];
  float* out = (float*)d_out;

  const int MA = in_sizes[0];        // M * A
  const int P  = in_sizes[2];        // pairs
  const int M  = out_size;           // molecules
  const unsigned A = (unsigned)(MA / M);
  // power-of-two fast path for the molecule index (A=64 -> shift 6)
  const int molShift = (A != 0u && (A & (A - 1u)) == 0u)
                           ? __builtin_ctz(A) : -1;

  float* ws = (float*)d_ws;
  // interleaved {cn, species} layout if scratch is big enough
  const bool ilv = ws_size >= (size_t)MA * sizeof(float2);

  int blocks0 = ((MA > M ? MA : M) + 255) / 256;

  int blocks1 = (P + 1023) / 1024;   // vec4: ~1 chunk per thread to start
  if (blocks1 > 4096) blocks1 = 4096;
  if (blocks1 < 1) blocks1 = 1;

  int blocks2 = (P + 2047) / 2048;   // ~8 pairs/thread at 256 threads
  if (blocks2 > 1024) blocks2 = 1024;
  if (blocks2 < 1) blocks2 = 1;

  size_t shbytes = (size_t)(NC * GG) * sizeof(float4) +
                   (size_t)(3 * NC * GG + 3 * NC + M) * sizeof(float);

  if (ilv) {
    k_init<true><<<blocks0, 256, 0, stream>>>(ws, MA, species, out, M);
    k_coordnum<true><<<blocks1, 256, 0, stream>>>(idx, dist, species, covalent,
                                                  ws, P);
    k_energy<true><<<blocks2, 256, shbytes, stream>>>(
        idx, dist, species, ws, c6tab, cnA, cnB, cutoff, sqq,
        a1, a2, s6, s8, out, P, A, molShift, M);
  } else {
    k_init<false><<<blocks0, 256, 0, stream>>>(ws, MA, species, out, M);
    k_coordnum<false><<<blocks1, 256, 0, stream>>>(idx, dist, species, covalent,
                                                   ws, P);
    k_energy<false><<<blocks2, 256, shbytes, stream>>>(
        idx, dist, species, ws, c6tab, cnA, cnB, cutoff, sqq,
        a1, a2, s6, s8, out, P, A, molShift, M);
  }
}